// NonMaxSuppression_54657753809178
// MI455X (gfx1250) — compile-verified
//
#include <hip/hip_runtime.h>
#include <hip/hip_bf16.h>
#include <stdint.h>

#define H 1024
#define W 1024
#define B 16
#define TOPK 4096
#define REP_THR 0.1f
#define TILE 16
#define HALO 2
#define TDIM (TILE + 2 * HALO)      // 20
#define TELEMS (TDIM * TDIM)        // 400
#define SORT_N 8192
#define SORT_PAIRS (SORT_N / 2)     // 4096
#define HBINS 8192                  // value-bit histogram bins
#define CAND2_CAP 32768             // filtered candidates per batch

__global__ void nms_zero_kernel(unsigned* p, unsigned nwords) {
    unsigned i = blockIdx.x * blockDim.x + threadIdx.x;
    unsigned stride = gridDim.x * blockDim.x;
    for (; i < nwords; i += stride) p[i] = 0u;
}

// Monotone bin of positive-float value bits; fine resolution over (0.5, 1].
__device__ __forceinline__ unsigned value_bin(unsigned bits) {
    if (bits < 0x3F000000u) return 0u;             // v < 0.5
    unsigned d = (bits - 0x3F000000u) >> 10;       // ~6.1e-5 wide bins
    return (d > HBINS - 2u) ? (HBINS - 1u) : (d + 1u);
}

// One 16x16 output tile per block (256 threads = 8 wave32).
// Stage 20x20 (tile+halo) into LDS with CDNA5 async global->LDS loads,
// then each thread tests its center pixel against the 5x5 window.
__global__ void __launch_bounds__(256)
nms_peak_kernel(const float* __restrict__ score,
                unsigned* __restrict__ cnt,
                unsigned long long* __restrict__ cand,
                unsigned cand_cap) {
    __shared__ float tile[TELEMS];

    const int b  = blockIdx.z;
    const int x0 = blockIdx.x * TILE;
    const int y0 = blockIdx.y * TILE;
    const int tid = threadIdx.x;

    const float* sb = score + (size_t)b * (size_t)(H * W);

    for (int i = tid; i < TELEMS; i += 256) {
        int ly = i / TDIM, lx = i % TDIM;
        int gy = y0 - HALO + ly;
        int gx = x0 - HALO + lx;
        bool ok = (gx >= 0) && (gx < W) && (gy >= 0) && (gy < H);
        if (ok) {
            unsigned lds_addr = (unsigned)(size_t)(void*)&tile[i]; // low 32b of flat LDS ptr = LDS offset
            unsigned goff = (unsigned)((gy * W + gx) * 4);
            asm volatile("global_load_async_to_lds_b32 %0, %1, %2"
                         :: "v"(lds_addr), "v"(goff), "s"(sb)
                         : "memory");
        }
    }
    asm volatile("s_wait_asynccnt 0" ::: "memory");
    for (int i = tid; i < TELEMS; i += 256) {
        int ly = i / TDIM, lx = i % TDIM;
        int gy = y0 - HALO + ly;
        int gx = x0 - HALO + lx;
        bool ok = (gx >= 0) && (gx < W) && (gy >= 0) && (gy < H);
        if (!ok) tile[i] = -__builtin_inff();
    }
    __syncthreads();

    const int lx = tid & (TILE - 1);
    const int ly = tid >> 4;
    const float v = tile[(ly + HALO) * TDIM + (lx + HALO)];

    float m = -__builtin_inff();
#pragma unroll
    for (int dy = 0; dy < 5; ++dy)
#pragma unroll
        for (int dx = 0; dx < 5; ++dx)
            m = fmaxf(m, tile[(ly + dy) * TDIM + (lx + dx)]);

    const bool peak = (v > REP_THR) && (v >= m);
    if (peak) {
        unsigned gx = (unsigned)(x0 + lx);
        unsigned gy = (unsigned)(y0 + ly);
        unsigned idx1d = gy * W + gx;
        unsigned long long key =
            ((unsigned long long)__float_as_uint(v) << 32) |
            (unsigned long long)(0xFFFFFFFFu - idx1d);
        unsigned slot = atomicAdd(&cnt[b], 1u);
        if (slot < cand_cap) cand[(size_t)b * cand_cap + slot] = key;
    }
}

// Per-batch histogram of candidate value bits.
__global__ void __launch_bounds__(1024)
nms_hist_kernel(const unsigned* __restrict__ cnt,
                const unsigned long long* __restrict__ cand,
                unsigned cand_cap,
                unsigned* __restrict__ hist) {
    const int b = blockIdx.x;
    unsigned n = cnt[b];
    if (n > cand_cap) n = cand_cap;
    const unsigned long long* cb = cand + (size_t)b * cand_cap;
    unsigned* hb = hist + (size_t)b * HBINS;
    for (unsigned i = threadIdx.x; i < n; i += 1024) {
        unsigned bits = (unsigned)(cb[i] >> 32);
        atomicAdd(&hb[value_bin(bits)], 1u);
    }
}

// Largest bin c with top-down cumulative count >= min(TOPK, n). Exact since
// every key in a lower bin is strictly smaller than every key in bins >= c.
__global__ void __launch_bounds__(1024)
nms_cutoff_kernel(const unsigned* __restrict__ cnt,
                  unsigned cand_cap,
                  const unsigned* __restrict__ hist,
                  unsigned* __restrict__ cutoff) {
    __shared__ unsigned seg[1024];
    const int b = blockIdx.x;
    const unsigned* hb = hist + (size_t)b * HBINS;
    const unsigned t = threadIdx.x;
    unsigned s = 0;
#pragma unroll
    for (int i = 0; i < HBINS / 1024; ++i) s += hb[t * (HBINS / 1024) + i];
    seg[t] = s;
    __syncthreads();
    if (t == 0) {
        unsigned n = cnt[b];
        if (n > cand_cap) n = cand_cap;
        unsigned k = (n < TOPK) ? n : TOPK;
        unsigned c = HBINS;  // select nothing when k == 0
        if (k > 0) {
            unsigned acc = 0;
            int sg = 1023;
            for (; sg >= 0; --sg) {
                if (acc + seg[sg] >= k) break;
                acc += seg[sg];
            }
            if (sg < 0) {
                c = 0;
            } else {
                int bin = sg * (HBINS / 1024) + (HBINS / 1024) - 1;
                for (; bin >= sg * (HBINS / 1024); --bin) {
                    acc += hb[bin];
                    if (acc >= k) break;
                }
                c = (bin < sg * (HBINS / 1024)) ? 0u : (unsigned)bin;
            }
        }
        cutoff[b] = c;
    }
}

// Compact keys with bin >= cutoff into the small filtered buffer.
__global__ void __launch_bounds__(1024)
nms_compact_kernel(const unsigned* __restrict__ cnt,
                   const unsigned long long* __restrict__ cand,
                   unsigned cand_cap,
                   const unsigned* __restrict__ cutoff,
                   unsigned* __restrict__ cnt2,
                   unsigned long long* __restrict__ cand2) {
    const int b = blockIdx.x;
    unsigned n = cnt[b];
    if (n > cand_cap) n = cand_cap;
    const unsigned c = cutoff[b];
    const unsigned long long* cb = cand + (size_t)b * cand_cap;
    unsigned long long* c2 = cand2 + (size_t)b * CAND2_CAP;
    for (unsigned i = threadIdx.x; i < n; i += 1024) {
        unsigned long long key = cb[i];
        if (value_bin((unsigned)(key >> 32)) >= c) {
            unsigned slot = atomicAdd(&cnt2[b], 1u);
            if (slot < CAND2_CAP) c2[slot] = key;
        }
    }
}

// One block per batch: load up to 8192 keys, bitonic sort descending; if more
// remain, stream 4096-key chunks into the high half and re-sort (exact merge).
__global__ void __launch_bounds__(1024)
nms_topk_kernel(const unsigned* __restrict__ cnt,
                const unsigned long long* __restrict__ cand,
                unsigned cand_cap,
                int* __restrict__ out) {
    __shared__ unsigned long long work[SORT_N]; // 64 KB

    const int b = blockIdx.x;
    const unsigned tid = threadIdx.x;

    unsigned n = cnt[b];
    if (n > cand_cap) n = cand_cap;
    const unsigned long long* cb = cand + (size_t)b * cand_cap;

    for (unsigned j = tid; j < SORT_N; j += 1024)
        work[j] = (j < n) ? cb[j] : 0ull;

    unsigned s = SORT_N;
    for (;;) {
        // Bitonic sort of 8192 keys, descending (key 0 = pad sentinel).
        for (unsigned k = 2; k <= SORT_N; k <<= 1) {
            for (unsigned jj = k >> 1; jj > 0; jj >>= 1) {
                __syncthreads();
                for (unsigned t = tid; t < SORT_PAIRS; t += 1024) {
                    unsigned i = ((t & ~(jj - 1)) << 1) | (t & (jj - 1));
                    unsigned l = i | jj;
                    unsigned long long a  = work[i];
                    unsigned long long bv = work[l];
                    bool up = ((i & k) == 0);
                    bool sw = up ? (a < bv) : (a > bv);
                    if (sw) { work[i] = bv; work[l] = a; }
                }
            }
        }
        __syncthreads();
        if (s >= n) break;
        for (unsigned j = tid; j < TOPK; j += 1024) {
            unsigned src = s + j;
            work[TOPK + j] = (src < n) ? cb[src] : 0ull;
        }
        s += TOPK;
        __syncthreads();
    }

    for (unsigned j = tid; j < TOPK; j += 1024) {
        unsigned long long key = work[j];
        int x = 0, y = 0;
        if (key != 0ull) {
            float v = __uint_as_float((unsigned)(key >> 32));
            if (v > REP_THR) {
                unsigned idx1d = 0xFFFFFFFFu - (unsigned)(key & 0xFFFFFFFFull);
                x = (int)(idx1d % W);
                y = (int)(idx1d / W);
            }
        }
        size_t o = ((size_t)b * TOPK + j) * 2;
        out[o + 0] = x;
        out[o + 1] = y;
    }
}

extern "C" void kernel_launch(void* const* d_in, const int* in_sizes, int n_in,
                              void* d_out, int out_size, void* d_ws, size_t ws_size,
                              hipStream_t stream) {
    const float* score = (const float*)d_in[0];
    int* out = (int*)d_out;
    char* ws = (char*)d_ws;

    // Workspace layout:
    //   [0,64)        cnt[16]
    //   [64,128)      cnt2[16]
    //   [128,192)     cutoff[16]
    //   [4096, +512K) hist[16][HBINS]
    //   then cand2[16][CAND2_CAP] (filtered, 4 MB)
    //   then cand[16][cand_cap]   (raw peak keys)
    unsigned* cnt    = (unsigned*)(ws + 0);
    unsigned* cnt2   = (unsigned*)(ws + 64);
    unsigned* cutoff = (unsigned*)(ws + 128);
    unsigned* hist   = (unsigned*)(ws + 4096);

    size_t off_c2   = 4096 + (size_t)B * HBINS * 4;          // 528,384
    size_t off_cand = off_c2 + (size_t)B * CAND2_CAP * 8;    // ~4.7 MB
    bool   filter   = ws_size >= off_cand + (size_t)B * 8 * 8192;
    size_t cand_off = filter ? off_cand : 4096;

    size_t cap = 1;
    if (ws_size > cand_off + 64) cap = (ws_size - cand_off) / ((size_t)B * 8ull);
    if (cap > 262144) cap = 262144;
    if (cap < 1) cap = 1;
    unsigned cand_cap = (unsigned)cap;

    unsigned long long* cand  = (unsigned long long*)(ws + cand_off);
    unsigned long long* cand2 = (unsigned long long*)(ws + off_c2);

    unsigned zero_words = filter ? (unsigned)(off_c2 / 4) : 1024u;
    nms_zero_kernel<<<(zero_words + 1023) / 1024, 1024, 0, stream>>>((unsigned*)ws, zero_words);

    dim3 gridA(W / TILE, H / TILE, B);
    nms_peak_kernel<<<gridA, 256, 0, stream>>>(score, cnt, cand, cand_cap);

    if (filter) {
        nms_hist_kernel<<<B, 1024, 0, stream>>>(cnt, cand, cand_cap, hist);
        nms_cutoff_kernel<<<B, 1024, 0, stream>>>(cnt, cand_cap, hist, cutoff);
        nms_compact_kernel<<<B, 1024, 0, stream>>>(cnt, cand, cand_cap, cutoff, cnt2, cand2);
        nms_topk_kernel<<<B, 1024, 0, stream>>>(cnt2, cand2, CAND2_CAP, out);
    } else {
        nms_topk_kernel<<<B, 1024, 0, stream>>>(cnt, cand, cand_cap, out);
    }
}